// RNN_74483322847433
// MI455X (gfx1250) — compile-verified
//
#include <hip/hip_runtime.h>
#include <hip/hip_bf16.h>
#include <math.h>

#define BB 16
#define TT 1024
#define VOCAB 8192
#define HID 1024

typedef __attribute__((ext_vector_type(2))) float v2f;
typedef __attribute__((ext_vector_type(4))) float v4f;
typedef __attribute__((ext_vector_type(8))) float v8f;

// ---------------- Kernel 1: embedding gather ----------------
// ws[t*B*H + b*H + h] = Wx_w[h*V + x[b*T+t]] + Wx_b[h]
__global__ __launch_bounds__(256) void embed_kernel(const int* __restrict__ x,
                                                    const float* __restrict__ Wxw,
                                                    const float* __restrict__ Wxb,
                                                    float* __restrict__ emb) {
    int idx = blockIdx.x * 256 + threadIdx.x;     // over T*B*H
    int hh = idx & (HID - 1);
    int tb = idx >> 10;                            // t*B + b
    int b  = tb & (BB - 1);
    int t  = tb >> 4;
    int tok = x[b * TT + t];
    emb[idx] = Wxw[(size_t)hh * VOCAB + tok] + Wxb[hh];
}

// ---------------- Kernel 2: persistent recurrence (1 WGP, 32 waves) ----------------
// hs[] holds emb on entry; overwritten in place with h_t per step.
__global__ __launch_bounds__(1024) void rnn_kernel(const float* __restrict__ h0,
                                                   const float* __restrict__ Wh,
                                                   float* __restrict__ hs,
                                                   float* __restrict__ hfin) {
    __shared__ float hbuf[2][BB][HID + 1];        // odd stride -> conflict-free A reads

    int tid = threadIdx.x;
    for (int i = tid; i < BB * HID; i += 1024)
        hbuf[0][i >> 10][i & (HID - 1)] = h0[i];
    __syncthreads();

    const int wave  = tid >> 5;
    const int lane  = tid & 31;
    const int ml    = lane & 15;                  // M row (batch) / N col within tile
    const int khalf = (lane >> 4) << 1;           // 0 or 2
    const int mbase = (lane >> 4) << 3;           // C/D row base: 0 or 8
    const int n0    = wave * 32;
    const int n     = n0 + ml;

    const float* whA = Wh + (size_t)n * HID;          // column n   of Wh^T == row n of Wh
    const float* whB = Wh + (size_t)(n + 16) * HID;   // column n+16

    int cur = 0;
    for (int t = 0; t < TT; ++t) {
        const float* hc = &hbuf[cur][0][0];
        v8f c0 = {}; v8f c1 = {};
        for (int k0 = 0; k0 < HID; k0 += 4) {
            int kk = k0 + khalf;
            v2f a;
            a.x = hc[ml * (HID + 1) + kk];
            a.y = hc[ml * (HID + 1) + kk + 1];
            v2f b0 = *(const v2f*)(whA + kk);
            v2f b1 = *(const v2f*)(whB + kk);
            c0 = __builtin_amdgcn_wmma_f32_16x16x4_f32(false, a, false, b0, (short)0, c0, false, false);
            c1 = __builtin_amdgcn_wmma_f32_16x16x4_f32(false, a, false, b1, (short)0, c1, false, false);
        }

        float* row = hs + (size_t)t * (BB * HID);     // emb[t] in, hs[t] out (same cells per wave)
        int nxt = cur ^ 1;
#pragma unroll
        for (int r = 0; r < 8; ++r) {
            int mm = mbase + r;                       // batch row
            float v0 = tanhf(c0[r] + row[mm * HID + n]);
            float v1 = tanhf(c1[r] + row[mm * HID + n + 16]);
            row[mm * HID + n]      = v0;
            row[mm * HID + n + 16] = v1;
            hbuf[nxt][mm][n]       = v0;
            hbuf[nxt][mm][n + 16]  = v1;
        }
        cur = nxt;
        __syncthreads();
    }

    // h_final from the last-written LDS buffer
    for (int i = tid; i < BB * HID; i += 1024)
        hfin[i] = hbuf[cur][i >> 10][i & (HID - 1)];
}

// ---------------- Kernel 3: output projection GEMM ----------------
// logits[b,t,v] = sum_h hs[t*B+b, h] * Wy[v,h] + Wyb[v]
// Block = 8 waves sharing one 16-row A tile staged in LDS; each wave owns 4 N-tiles.
#define ASTRIDE (HID + 4)   // 1028: rows 16B-aligned, conflict-free dual-half-wave reads

__global__ __launch_bounds__(256) void logits_kernel(const float* __restrict__ hs,
                                                     const float* __restrict__ Wy,
                                                     const float* __restrict__ Wyb,
                                                     float* __restrict__ out) {
    __shared__ float atile[BB][ASTRIDE];

    const int tid  = threadIdx.x;
    const int m0   = blockIdx.x * 16;                    // M tile over T*B (16 consecutive rows)

    // cooperative coalesced copy: 64KB contiguous region hs[m0*HID .. ) -> LDS
    {
        const float* src = hs + (size_t)m0 * HID;
#pragma unroll
        for (int c = 0; c < (BB * HID) / (256 * 4); ++c) {   // 16 iterations of float4
            int g   = (c * 256 + tid) * 4;                   // flat float index
            int row = g >> 10;
            int col = g & (HID - 1);
            v4f v = *(const v4f*)(src + g);
            *(v4f*)&atile[row][col] = v;
        }
    }
    __syncthreads();

    const int wave  = tid >> 5;
    const int lane  = tid & 31;
    const int ml    = lane & 15;
    const int khalf = (lane >> 4) << 1;
    const int mbase = (lane >> 4) << 3;
    const int n0    = blockIdx.y * 512 + wave * 64;       // 4x16 N tiles per wave

    const float* b0r = Wy + (size_t)(n0 + ml) * HID;
    const float* b1r = b0r + (size_t)16 * HID;
    const float* b2r = b0r + (size_t)32 * HID;
    const float* b3r = b0r + (size_t)48 * HID;

    v8f c0 = {}, c1 = {}, c2 = {}, c3 = {};
    for (int k0 = 0; k0 < HID; k0 += 4) {
        int kk = k0 + khalf;
        v2f a;
        a.x = atile[ml][kk];
        a.y = atile[ml][kk + 1];
        v2f b0 = *(const v2f*)(b0r + kk);
        v2f b1 = *(const v2f*)(b1r + kk);
        v2f b2 = *(const v2f*)(b2r + kk);
        v2f b3 = *(const v2f*)(b3r + kk);
        c0 = __builtin_amdgcn_wmma_f32_16x16x4_f32(false, a, false, b0, (short)0, c0, false, false);
        c1 = __builtin_amdgcn_wmma_f32_16x16x4_f32(false, a, false, b1, (short)0, c1, false, false);
        c2 = __builtin_amdgcn_wmma_f32_16x16x4_f32(false, a, false, b2, (short)0, c2, false, false);
        c3 = __builtin_amdgcn_wmma_f32_16x16x4_f32(false, a, false, b3, (short)0, c3, false, false);
    }

    const int nn = n0 + ml;
    const float bias0 = Wyb[nn], bias1 = Wyb[nn + 16], bias2 = Wyb[nn + 32], bias3 = Wyb[nn + 48];
#pragma unroll
    for (int r = 0; r < 8; ++r) {
        int mrow = m0 + mbase + r;                // = t*B + b
        int tt = mrow >> 4;
        int bb = mrow & 15;
        float* o = out + ((size_t)bb * TT + tt) * VOCAB + nn;
        o[0]  = c0[r] + bias0;
        o[16] = c1[r] + bias1;
        o[32] = c2[r] + bias2;
        o[48] = c3[r] + bias3;
    }
}

extern "C" void kernel_launch(void* const* d_in, const int* in_sizes, int n_in,
                              void* d_out, int out_size, void* d_ws, size_t ws_size,
                              hipStream_t stream) {
    (void)in_sizes; (void)n_in; (void)out_size; (void)ws_size;
    const int*   x   = (const int*)d_in[0];
    const float* h0  = (const float*)d_in[1];
    const float* Wxw = (const float*)d_in[2];
    const float* Wxb = (const float*)d_in[3];
    const float* Whw = (const float*)d_in[4];
    const float* Wyw = (const float*)d_in[5];
    const float* Wyb = (const float*)d_in[6];

    float* out  = (float*)d_out;
    float* hs   = (float*)d_ws;                                // [T,B,H] = 64 MB scratch
    float* hfin = out + (size_t)BB * TT * VOCAB;               // h_final after logits

    embed_kernel<<<(TT * BB * HID) / 256, 256, 0, stream>>>(x, Wxw, Wxb, hs);
    rnn_kernel<<<1, 1024, 0, stream>>>(h0, Whw, hs, hfin);
    logits_kernel<<<dim3((TT * BB) / 16, VOCAB / 512), 256, 0, stream>>>(hs, Wyw, Wyb, out);
}